// Sparsemax_3693671875246
// MI455X (gfx1250) — compile-verified
//
#include <hip/hip_runtime.h>
#include <hip/hip_bf16.h>

// Sparsemax over rows of an 8192-wide f32 matrix, one workgroup per row.
// Single HBM pass: row -> LDS via gfx1250 async-to-LDS, threshold search
// entirely in registers/LDS (bisection + exact Michelot refinement),
// result -> HBM via async-from-LDS.

#define D      8192               // row length
#define T      256                // threads per block (8 wave32)
#define PER    (D / T)            // 32 elements per thread
#define CHUNKS (D * 4 / 16 / T)   // 8 x 16B chunks per thread
#define NW     (T / 32)           // 8 waves per block

#define AS1 __attribute__((address_space(1)))
#define AS3 __attribute__((address_space(3)))

typedef int v4i __attribute__((ext_vector_type(4)));

#if __has_builtin(__builtin_amdgcn_global_load_async_to_lds_b128) && \
    __has_builtin(__builtin_amdgcn_global_store_async_from_lds_b128) && \
    __has_builtin(__builtin_amdgcn_s_wait_asynccnt)
#define USE_ASYNC 1
#else
#define USE_ASYNC 0
#endif

__global__ __launch_bounds__(T) void sparsemax8192_kernel(
    const float* __restrict__ x, float* __restrict__ y) {
  __shared__ float row[D];
  __shared__ float rbuf[2 * NW];

  const int t = threadIdx.x;
  const size_t base = (size_t)blockIdx.x * (size_t)D;

  // ---- Stage 1: stream row into LDS (async data mover, bypasses VGPRs) ----
#if USE_ASYNC
  {
    const v4i* g = (const v4i*)(x + base);
    v4i* l = (v4i*)row;
#pragma unroll
    for (int i = 0; i < CHUNKS; ++i) {
      const int c = t + i * T;  // consecutive lanes -> consecutive 16B chunks
      __builtin_amdgcn_global_load_async_to_lds_b128(
          (AS1 v4i*)(g + c), (AS3 v4i*)(l + c), 0, 0);
    }
    __builtin_amdgcn_s_wait_asynccnt(0);
  }
#else
  {
    const float4* g = (const float4*)(x + base);
    float4* l = (float4*)row;
#pragma unroll
    for (int i = 0; i < CHUNKS; ++i) {
      const int c = t + i * T;
      l[c] = g[c];
    }
  }
#endif
  __syncthreads();

  // ---- Stage 2: strided row slice into registers (bank-conflict-free) ----
  float v[PER];
#pragma unroll
  for (int j = 0; j < PER; ++j) v[j] = row[t + j * T];

  // ---- Stage 3: block max -> bisection bracket [m-1, m) ----
  float m = v[0];
#pragma unroll
  for (int j = 1; j < PER; ++j) m = fmaxf(m, v[j]);
#pragma unroll
  for (int o = 16; o > 0; o >>= 1) m = fmaxf(m, __shfl_xor(m, o, 32));
  if ((t & 31) == 0) rbuf[t >> 5] = m;
  __syncthreads();
  float mx = rbuf[0];
#pragma unroll
  for (int w = 1; w < NW; ++w) mx = fmaxf(mx, rbuf[w]);
  __syncthreads();

  // ---- Stage 4: bisection on f(tau) = sum(relu(x - tau)) - 1 ----
  float lo = mx - 1.0f;
  float hi = mx;
#pragma unroll 1
  for (int it = 0; it < 12; ++it) {
    const float tau = 0.5f * (lo + hi);
    float s = 0.0f;
#pragma unroll
    for (int j = 0; j < PER; ++j) s += fmaxf(v[j] - tau, 0.0f);
#pragma unroll
    for (int o = 16; o > 0; o >>= 1) s += __shfl_xor(s, o, 32);
    if ((t & 31) == 0) rbuf[t >> 5] = s;
    __syncthreads();
    float tot = 0.0f;
#pragma unroll
    for (int w = 0; w < NW; ++w) tot += rbuf[w];  // identical order in all lanes
    __syncthreads();
    if (tot > 1.0f) lo = tau; else hi = tau;      // block-uniform branch
  }

  // ---- Stage 5: exact refinement, tau = (sum(top-k) - 1) / k ----
  float tau = 0.5f * (lo + hi);
#pragma unroll 1
  for (int it = 0; it < 8; ++it) {
    float k = 0.0f, s = 0.0f;
#pragma unroll
    for (int j = 0; j < PER; ++j) {
      if (v[j] > tau) { k += 1.0f; s += v[j]; }
    }
#pragma unroll
    for (int o = 16; o > 0; o >>= 1) {
      k += __shfl_xor(k, o, 32);
      s += __shfl_xor(s, o, 32);
    }
    if ((t & 31) == 0) { rbuf[t >> 5] = k; rbuf[NW + (t >> 5)] = s; }
    __syncthreads();
    float K = 0.0f, S = 0.0f;
#pragma unroll
    for (int w = 0; w < NW; ++w) { K += rbuf[w]; S += rbuf[NW + w]; }
    __syncthreads();
    K = fmaxf(K, 1.0f);
    const float tn = (S - 1.0f) / K;
    if (tn == tau) break;  // block-uniform fixed point -> exact threshold
    tau = tn;
  }

  // ---- Stage 6: emit max(x - tau, 0) through LDS, async store to HBM ----
#pragma unroll
  for (int j = 0; j < PER; ++j)
    row[t + j * T] = fmaxf(v[j] - tau, 0.0f);
  __syncthreads();

#if USE_ASYNC
  {
    v4i* g = (v4i*)(y + base);
    const v4i* l = (const v4i*)row;
#pragma unroll
    for (int i = 0; i < CHUNKS; ++i) {
      const int c = t + i * T;
      __builtin_amdgcn_global_store_async_from_lds_b128(
          (AS1 v4i*)(g + c), (AS3 v4i*)(l + c), 0, 0);
    }
    __builtin_amdgcn_s_wait_asynccnt(0);
  }
#else
  {
    float4* g = (float4*)(y + base);
    const float4* l = (const float4*)row;
#pragma unroll
    for (int i = 0; i < CHUNKS; ++i) {
      const int c = t + i * T;
      g[c] = l[c];
    }
  }
#endif
}

extern "C" void kernel_launch(void* const* d_in, const int* in_sizes, int n_in,
                              void* d_out, int out_size, void* d_ws, size_t ws_size,
                              hipStream_t stream) {
  (void)n_in; (void)out_size; (void)d_ws; (void)ws_size;
  const float* x = (const float*)d_in[0];
  float* y = (float*)d_out;
  const int rows = in_sizes[0] / D;  // 8192 rows, one block per row
  sparsemax8192_kernel<<<dim3(rows), dim3(T), 0, stream>>>(x, y);
}